// GridBatchSampler_65309272703432
// MI455X (gfx1250) — compile-verified
//
#include <hip/hip_runtime.h>
#include <stdint.h>
#include <limits.h>

// ---------------- problem constants ----------------
#define B_      32
#define N_      131072
#define BN_     (B_ * N_)          // 4194304
#define T_      32768
#define CELL    0.05f
// radix sort config: 5-bit digits, 6 passes (keys < 2^30)
#define RBITS   5
#define RADIX   32
#define NPASS   6
#define CHUNK   4096               // elements per block (256 thr x 16)
#define NBLK    (BN_ / CHUNK)      // 1024 blocks exactly

// ---------------- TDM (gfx1250 Tensor Data Mover) ----------------
#if defined(__gfx1250__) && __has_builtin(__builtin_amdgcn_tensor_load_to_lds)
#define USE_TDM 1
#else
#define USE_TDM 0
#endif

#if USE_TDM
typedef unsigned int u32x4 __attribute__((ext_vector_type(4)));
typedef int          i32x8 __attribute__((ext_vector_type(8)));
typedef int          i32x4 __attribute__((ext_vector_type(4)));

// DMA CHUNK dwords from global to LDS via the TDM (1-row 2D tile).
__device__ __forceinline__ void tdm_load_chunk(void* ldsDst, const void* gsrc) {
  unsigned ldsOff = (unsigned)(uintptr_t)ldsDst;               // low 32b = LDS offset
  unsigned long long ga = (unsigned long long)(uintptr_t)gsrc;
  u32x4 g0;
  g0[0] = 1u;                                                  // count=1, user desc
  g0[1] = ldsOff;                                              // lds_addr
  g0[2] = (unsigned)ga;                                        // global_addr[31:0]
  g0[3] = (unsigned)((ga >> 32) & 0x01FFFFFFull) | (2u << 30); // addr[56:32] | type=2
  i32x8 g1;
  g1[0] = 0x20000;                          // data_size=2 (4B), no mask/barrier/pad
  g1[1] = (int)((CHUNK & 0xFFFF) << 16);    // tensor_dim0 lo16 -> bits 63:48
  g1[2] = (CHUNK >> 16) | (1 << 16);        // tensor_dim0 hi16 | tensor_dim1 lo16 (=1)
  g1[3] = (int)(CHUNK << 16);               // tensor_dim1 hi16 | tile_dim0 (=CHUNK)
  g1[4] = 1;                                // tile_dim1=1, tile_dim2=0
  g1[5] = CHUNK;                            // tensor_dim0_stride lo32
  g1[6] = 0;
  g1[7] = 0;
  i32x4 z4 = {0, 0, 0, 0};
  i32x8 z8 = {0, 0, 0, 0, 0, 0, 0, 0};
  __builtin_amdgcn_tensor_load_to_lds(g0, g1, z4, z4, z8, 0);
}
#endif

// ---------------- kernels ----------------

// init: ul_idx_inv <- sentinel BN ; min/max seeds
__global__ void kInit(int* __restrict__ ulInv, int* __restrict__ gmm) {
  int i = blockIdx.x * blockDim.x + threadIdx.x;
  if (i < BN_) ulInv[i] = BN_;
  if (i == 0) { gmm[0] = INT_MAX; gmm[1] = INT_MIN; }
}

// global min/max of quantized grid indices over all 3*BN floats
__global__ void kMinMax(const float* __restrict__ x, int* __restrict__ gmm) {
  __shared__ int smin[256];
  __shared__ int smax[256];
  int tid = blockIdx.x * blockDim.x + threadIdx.x;
  int stride = gridDim.x * blockDim.x;
  int mn = INT_MAX, mx = INT_MIN;
  for (int i = tid; i < 3 * BN_; i += stride) {
    int g = (int)rintf(x[i] / CELL);
    mn = min(mn, g);
    mx = max(mx, g);
  }
  smin[threadIdx.x] = mn; smax[threadIdx.x] = mx;
  __syncthreads();
  for (int off = 128; off > 0; off >>= 1) {
    if (threadIdx.x < off) {
      smin[threadIdx.x] = min(smin[threadIdx.x], smin[threadIdx.x + off]);
      smax[threadIdx.x] = max(smax[threadIdx.x], smax[threadIdx.x + off]);
    }
    __syncthreads();
  }
  if (threadIdx.x == 0) {
    atomicMin(&gmm[0], smin[0]);
    atomicMax(&gmm[1], smax[0]);
  }
}

// build linear voxel keys (int32, >=0) + identity payload
__global__ void kKeys(const float* __restrict__ x, const int* __restrict__ gmm,
                      int* __restrict__ keys, int* __restrict__ ord) {
  int i = blockIdx.x * blockDim.x + threadIdx.x;
  if (i >= BN_) return;
  __builtin_prefetch(x + (size_t)i * 3 + 2048, 0, 0);
  int gmin = gmm[0];
  int gs = gmm[1] - gmin + 2;
  int g0 = (int)rintf(x[3 * (size_t)i + 0] / CELL) - gmin;
  int g1 = (int)rintf(x[3 * (size_t)i + 1] / CELL) - gmin;
  int g2 = (int)rintf(x[3 * (size_t)i + 2] / CELL) - gmin;
  int b = i / N_;
  int key = gs * (gs * g0 + g1) + g2 + gs * gs * gs * b;
  keys[i] = key;
  ord[i] = i;
}

// radix pass 1/3: per-block 32-bin histogram (LDS ds_add atomics)
__global__ void kCount(const int* __restrict__ keys, int* __restrict__ counts, int shift) {
  __shared__ int hist[RADIX];
  int t = threadIdx.x, blk = blockIdx.x;
  if (t < RADIX) hist[t] = 0;
  __syncthreads();
  int base = blk * CHUNK + t * 16;
#pragma unroll
  for (int j = 0; j < 16; ++j) {
    int d = (keys[base + j] >> shift) & (RADIX - 1);
    atomicAdd(&hist[d], 1);
  }
  __syncthreads();
  if (t < RADIX) counts[t * NBLK + blk] = hist[t];
}

// radix pass 2/3: turn counts[d][blk] into global exclusive offsets, in place.
// one block of NBLK(=1024) threads.
__global__ void kScanDigits(int* __restrict__ counts) {
  __shared__ int sbuf[NBLK];
  __shared__ int stot[RADIX];
  __shared__ int sbase[RADIX];
  int t = threadIdx.x;
  for (int d = 0; d < RADIX; ++d) {
    int v = counts[d * NBLK + t];
    sbuf[t] = v;
    __syncthreads();
    for (int off = 1; off < NBLK; off <<= 1) {
      int a = (t >= off) ? sbuf[t - off] : 0;
      __syncthreads();
      sbuf[t] += a;
      __syncthreads();
    }
    if (t == NBLK - 1) stot[d] = sbuf[t];
    counts[d * NBLK + t] = sbuf[t] - v;   // exclusive within digit
    __syncthreads();
  }
  if (t == 0) {
    int run = 0;
    for (int d = 0; d < RADIX; ++d) { sbase[d] = run; run += stot[d]; }
  }
  __syncthreads();
  for (int d = 0; d < RADIX; ++d)
    counts[d * NBLK + t] += sbase[d];
}

// radix pass 3/3: stable scatter. Chunk staged into LDS via TDM.
__global__ void kScatter(const int* __restrict__ keysIn, const int* __restrict__ ordIn,
                         int* __restrict__ keysOut, int* __restrict__ ordOut,
                         const int* __restrict__ offs, int shift) {
  __shared__ int lk[CHUNK];
  __shared__ int lo[CHUNK];
  __shared__ int mat[RADIX][256];
  __shared__ int sb[256];
  int t = threadIdx.x, blk = blockIdx.x;

#if USE_TDM
  if (t < 32) {  // one wave issues the DMA; TDM ignores EXEC, wave-level issue
    tdm_load_chunk(lk, keysIn + (size_t)blk * CHUNK);
    tdm_load_chunk(lo, ordIn + (size_t)blk * CHUNK);
    __builtin_amdgcn_s_wait_tensorcnt(0);
  }
#else
  for (int j = t; j < CHUNK; j += 256) {
    lk[j] = keysIn[(size_t)blk * CHUNK + j];
    lo[j] = ordIn[(size_t)blk * CHUNK + j];
  }
#endif
  // zero rank matrix while DMA is in flight
  for (int d = 0; d < RADIX; ++d) mat[d][t] = 0;
  __syncthreads();

  int myd[16];
#pragma unroll
  for (int j = 0; j < 16; ++j) {
    myd[j] = (lk[t * 16 + j] >> shift) & (RADIX - 1);
    mat[myd[j]][t]++;                 // own column: no race
  }
  __syncthreads();

  // per-digit exclusive scan across threads (thread-minor => stability)
  for (int d = 0; d < RADIX; ++d) {
    int v = mat[d][t];
    sb[t] = v;
    __syncthreads();
    for (int off = 1; off < 256; off <<= 1) {
      int a = (t >= off) ? sb[t - off] : 0;
      __syncthreads();
      sb[t] += a;
      __syncthreads();
    }
    mat[d][t] = sb[t] - v;
    __syncthreads();
  }

#pragma unroll
  for (int j = 0; j < 16; ++j) {
    int d = myd[j];
    int run = 0;
#pragma unroll
    for (int j2 = 0; j2 < 16; ++j2)
      if (j2 < j && myd[j2] == d) run++;
    int pos = offs[d * NBLK + blk] + mat[d][t] + run;
    keysOut[pos] = lk[t * 16 + j];
    ordOut[pos]  = lo[t * 16 + j];
  }
}

// ---------- generic 4M-element exclusive scans (3-kernel) ----------
// MODE 0: input = is_new flag derived from sorted keys; MODE 1: input = int array
template <int MODE>
__global__ void kS1(const int* __restrict__ in, int* __restrict__ out,
                    int* __restrict__ bsums) {
  __shared__ int sb[256];
  int t = threadIdx.x, blk = blockIdx.x;
  int base = blk * CHUNK + t * 16;
  int loc[16];
  int s = 0;
#pragma unroll
  for (int j = 0; j < 16; ++j) {
    int i = base + j;
    int v;
    if (MODE == 0) v = (i == 0) ? 1 : (in[i] != in[i - 1]);
    else           v = in[i];
    loc[j] = s;
    s += v;
  }
  sb[t] = s;
  __syncthreads();
  for (int off = 1; off < 256; off <<= 1) {
    int a = (t >= off) ? sb[t - off] : 0;
    __syncthreads();
    sb[t] += a;
    __syncthreads();
  }
  int thrExcl = sb[t] - s;
#pragma unroll
  for (int j = 0; j < 16; ++j) out[base + j] = thrExcl + loc[j];
  if (t == 255) bsums[blk] = sb[255];
}

__global__ void kS2(int* __restrict__ bsums) {  // 1 block, NBLK threads
  __shared__ int sb[NBLK];
  int t = threadIdx.x;
  int v = bsums[t];
  sb[t] = v;
  __syncthreads();
  for (int off = 1; off < NBLK; off <<= 1) {
    int a = (t >= off) ? sb[t - off] : 0;
    __syncthreads();
    sb[t] += a;
    __syncthreads();
  }
  bsums[t] = sb[t] - v;  // exclusive block offsets
}

template <int MODE>
__global__ void kS3(const int* __restrict__ keys, int* __restrict__ arr,
                    const int* __restrict__ bsums) {
  int i = blockIdx.x * blockDim.x + threadIdx.x;
  if (i >= BN_) return;
  int off = bsums[i / CHUNK];
  if (MODE == 0) {  // arr becomes inclusive-cumsum(is_new)-1  (segment id)
    int v = (i == 0) ? 1 : (keys[i] != keys[i - 1]);
    arr[i] = arr[i] + off + v - 1;
  } else {          // arr becomes global exclusive prefix
    arr[i] += off;
  }
}

// mark first occurrences (original order) + record first position per segment
__global__ void kMarkFirst(const int* __restrict__ keys, const int* __restrict__ ord,
                           const int* __restrict__ segId, int* __restrict__ maskO,
                           int* __restrict__ firstPos) {
  int s = blockIdx.x * blockDim.x + threadIdx.x;
  if (s >= BN_) return;
  int isn = (s == 0) ? 1 : (keys[s] != keys[s - 1]);
  int o = ord[s];
  maskO[o] = isn;
  if (isn) firstPos[segId[s]] = o;
}

// ul_idx: appearance-order unique id per point
__global__ void kUlIdx(const int* __restrict__ ord, const int* __restrict__ segId,
                       const int* __restrict__ firstPos, const int* __restrict__ pmE,
                       int* __restrict__ ulIdx) {
  int s = blockIdx.x * blockDim.x + threadIdx.x;
  if (s >= BN_) return;
  int o = ord[s];
  ulIdx[o] = pmE[firstPos[segId[s]]];
}

// outputs: stable per-batch top-k of mask + ul_idx_inv scatter
__global__ void kOut(const float* __restrict__ x, const int* __restrict__ maskO,
                     const int* __restrict__ pmE, float* __restrict__ ySel,
                     int* __restrict__ idxOut, float* __restrict__ maskSel,
                     int* __restrict__ ulInv) {
  int i = blockIdx.x * blockDim.x + threadIdx.x;
  if (i >= BN_) return;
  int b = i / N_;
  int local = i - b * N_;
  int m = maskO[i];
  int pmStart = pmE[b * N_];
  int last = b * N_ + N_ - 1;
  int cb = pmE[last] + maskO[last] - pmStart;       // uniques in this batch
  int onesRank = pmE[i] - pmStart;
  int slot = m ? onesRank : (cb + (local - onesRank));
  if (slot < T_) {
    size_t o = (size_t)b * T_ + slot;
#pragma unroll
    for (int c = 0; c < 3; ++c) {
      float v = x[3 * (size_t)i + c];
      ySel[o * 3 + c] = CELL * rintf(v / CELL);
    }
    idxOut[o * 2 + 0] = b;
    idxOut[o * 2 + 1] = local;
    maskSel[o] = m ? 1.0f : 0.0f;
  }
  if (m) ulInv[pmE[i]] = i;   // sorted first-occurrence positions
}

// ---------------- launcher ----------------
extern "C" void kernel_launch(void* const* d_in, const int* in_sizes, int n_in,
                              void* d_out, int out_size, void* d_ws, size_t ws_size,
                              hipStream_t stream) {
  (void)in_sizes; (void)n_in; (void)out_size; (void)ws_size;
  const float* x = (const float*)d_in[0];

  // output layout (flat, return order)
  float* ySel   = (float*)d_out;                 // 32*32768*3
  int*   idxOut = (int*)(ySel + (size_t)B_ * T_ * 3);   // 32*32768*2
  float* maskSel= (float*)(idxOut + (size_t)B_ * T_ * 2);
  int*   ulIdx  = (int*)(maskSel + (size_t)B_ * T_);    // BN
  int*   ulInv  = ulIdx + BN_;                          // BN

  // workspace layout (int32 units)
  int* w = (int*)d_ws;
  int* keysA  = w;
  int* keysB  = w + (size_t)BN_;
  int* ordA   = w + 2 * (size_t)BN_;
  int* ordB   = w + 3 * (size_t)BN_;
  int* maskO  = w + 4 * (size_t)BN_;
  int* pmE    = w + 5 * (size_t)BN_;
  int* segId  = keysB;     // reuse after sort (sort ends in A)
  int* firstPos = ordB;    // reuse after sort
  int* counts = w + 6 * (size_t)BN_;            // RADIX*NBLK = 32768
  int* bsums  = counts + RADIX * NBLK;          // NBLK
  int* gmm    = bsums + NBLK;                   // min,max

  const int PB = BN_ / 256;  // 16384 per-element blocks

  kInit<<<PB, 256, 0, stream>>>(ulInv, gmm);
  kMinMax<<<2048, 256, 0, stream>>>(x, gmm);
  kKeys<<<PB, 256, 0, stream>>>(x, gmm, keysA, ordA);

  int* kIn = keysA; int* kOut_ = keysB;
  int* oIn = ordA;  int* oOut  = ordB;
  for (int p = 0; p < NPASS; ++p) {
    int shift = p * RBITS;
    kCount<<<NBLK, 256, 0, stream>>>(kIn, counts, shift);
    kScanDigits<<<1, NBLK, 0, stream>>>(counts);
    kScatter<<<NBLK, 256, 0, stream>>>(kIn, oIn, kOut_, oOut, counts, shift);
    int* tk = kIn; kIn = kOut_; kOut_ = tk;
    int* to = oIn; oIn = oOut;  oOut  = to;
  }
  // NPASS even -> sorted keys/order back in keysA/ordA

  // segment ids over sorted keys
  kS1<0><<<NBLK, 256, 0, stream>>>(keysA, segId, bsums);
  kS2<<<1, NBLK, 0, stream>>>(bsums);
  kS3<0><<<PB, 256, 0, stream>>>(keysA, segId, bsums);

  kMarkFirst<<<PB, 256, 0, stream>>>(keysA, ordA, segId, maskO, firstPos);

  // exclusive prefix of mask in original order
  kS1<1><<<NBLK, 256, 0, stream>>>(maskO, pmE, bsums);
  kS2<<<1, NBLK, 0, stream>>>(bsums);
  kS3<1><<<PB, 256, 0, stream>>>(maskO, pmE, bsums);

  kUlIdx<<<PB, 256, 0, stream>>>(ordA, segId, firstPos, pmE, ulIdx);
  kOut<<<PB, 256, 0, stream>>>(x, maskO, pmE, ySel, idxOut, maskSel, ulInv);
}